// VQLayer_57741540327930
// MI455X (gfx1250) — compile-verified
//
#include <hip/hip_runtime.h>

typedef __attribute__((ext_vector_type(2))) float v2f;
typedef __attribute__((ext_vector_type(8))) float v8f;

#define B_ 8
#define N_ 4096
#define D_ 128
#define K_ 1024
#define KT 64            // K columns staged in LDS per tile
#define TSTRIDE 134      // transposed tile row stride (words): 128 + 6 -> stride = 6 mod 64 banks
#define TILE_WORDS (KT * TSTRIDE)
#define ROWS_PER_WG 128

// ---------- kernel 1: per-code squared norms  eNorm[b,k] = sum_d e[b,d,k]^2 ----------
__global__ void vq_enorm(const float* __restrict__ E, float* __restrict__ eNorm) {
    int idx = blockIdx.x * blockDim.x + threadIdx.x;  // b*K + k
    int b = idx >> 10;
    int k = idx & (K_ - 1);
    const float* ep = E + (size_t)b * D_ * K_ + k;
    float s = 0.f;
#pragma unroll 8
    for (int d = 0; d < D_; ++d) {
        float v = ep[(size_t)d * K_];
        s += v * v;
    }
    eNorm[idx] = s;
}

// ---------- kernel 2: fused  score = ||e||^2 - 2 x.e  via f32 WMMA, argmin, gather, loss ----------
__global__ __launch_bounds__(256)
void vq_main(const float* __restrict__ X, const float* __restrict__ E,
             const float* __restrict__ eNorm, float* __restrict__ out,
             float* __restrict__ partial) {
    // double-buffered transposed tiles: smem[buf][k'][d], row stride 134 words (2 x ~33.5 KB)
    extern __shared__ float smem[];
    __shared__ int   idx_s[8][16];
    __shared__ float wsum[8];

    const int tid  = threadIdx.x;
    const int w    = tid >> 5;      // wave id (8 waves)
    const int l    = tid & 31;      // lane
    const int lm   = l & 15;        // M row (A) / N col (B,C)
    const int koff = (l >> 4) * 2;  // K split across lane halves (ISA 7.12.2)

    const int wgPerBatch = N_ / ROWS_PER_WG;  // 32
    const int b  = blockIdx.x / wgPerBatch;
    const int n0 = (blockIdx.x % wgPerBatch) * ROWS_PER_WG;
    const int rowBase = n0 + w * 16;

    const float* ebase = E + (size_t)b * D_ * K_;

    // A fragments: 16 rows x 128 d, scaled by -2 (WMMA accumulates -2*x.e; ||e||^2 added at epilogue)
    // lanes 0-15 hold M=lane with K=dd,dd+1; lanes 16-31 hold M=lane-16 with K=dd+2,dd+3
    v2f a_frag[32];
    {
        const float* xrow = X + ((size_t)b * N_ + rowBase + lm) * D_;
#pragma unroll
        for (int d4 = 0; d4 < 32; ++d4) {
            float2 t = *(const float2*)(xrow + 4 * d4 + koff);
            a_frag[d4].x = -2.0f * t.x;
            a_frag[d4].y = -2.0f * t.y;
        }
    }

    // staging regs: 2048 float4 per tile, 8 per thread
    float4 stg[8];
    int srow[8], sc4[8];
#pragma unroll
    for (int i = 0; i < 8; ++i) {
        int lin = tid + 256 * i;   // 0..2047
        srow[i] = lin >> 4;        // d row (0..127)
        sc4[i]  = lin & 15;        // float4 index within the 64-wide k tile
    }

    // prologue: load + transpose-store tile 0 into buffer 0
#pragma unroll
    for (int i = 0; i < 8; ++i)
        stg[i] = *(const float4*)(ebase + (size_t)srow[i] * K_ + 4 * sc4[i]);
#pragma unroll
    for (int i = 0; i < 8; ++i) {
        float v4[4] = {stg[i].x, stg[i].y, stg[i].z, stg[i].w};
#pragma unroll
        for (int j = 0; j < 4; ++j)
            smem[(4 * sc4[i] + j) * TSTRIDE + srow[i]] = v4[j];
    }

    float minval[8];
    int   minidx[8];
#pragma unroll
    for (int r = 0; r < 8; ++r) { minval[r] = 3.4e38f; minidx[r] = 0; }

    for (int kt = 0; kt < K_; kt += KT) {
        // prefetch next tile from global while computing this one
        if (kt + KT < K_) {
#pragma unroll
            for (int i = 0; i < 8; ++i)
                stg[i] = *(const float4*)(ebase + (size_t)srow[i] * K_ + (kt + KT) + 4 * sc4[i]);
        }

        __syncthreads();   // stores to current buffer (prev iter / prologue) complete
        const float* tile = smem + (((kt >> 6) & 1) ? TILE_WORDS : 0);

#pragma unroll
        for (int sub = 0; sub < KT / 16; ++sub) {
            const int k = kt + sub * 16 + lm;          // this lane's code column
            const float en = eNorm[b * K_ + k];        // latency hidden behind the WMMA chain
            v8f acc = {};                              // SRC2 = inline 0, no seeding movs

            // B fragment = one contiguous 8-byte LDS read: (e[dd+koff][k'], e[dd+koff+1][k'])
            const float* bp = &tile[(sub * 16 + lm) * TSTRIDE + koff];
#pragma unroll
            for (int d4 = 0; d4 < 32; ++d4) {
                v2f bf = *(const v2f*)(bp + 4 * d4);   // ds_load_b64, imm offsets 16B apart
                acc = __builtin_amdgcn_wmma_f32_16x16x4_f32(
                        false, a_frag[d4], false, bf, (short)0, acc, false, false);
            }
            // score = ||e||^2 - 2 x.e ; acc[r]: row M=r (lanes 0-15) / M=r+8 (lanes 16-31), col k
#pragma unroll
            for (int r = 0; r < 8; ++r) {
                float s = acc[r] + en;
                if (s < minval[r]) { minval[r] = s; minidx[r] = k; }
            }
        }

        // transpose-store next tile into the other buffer (other waves may still read this one)
        if (kt + KT < K_) {
            float* nxt = smem + (((kt >> 6) & 1) ? 0 : TILE_WORDS);
#pragma unroll
            for (int i = 0; i < 8; ++i) {
                float v4[4] = {stg[i].x, stg[i].y, stg[i].z, stg[i].w};
#pragma unroll
                for (int j = 0; j < 4; ++j)
                    nxt[(4 * sc4[i] + j) * TSTRIDE + srow[i]] = v4[j];
            }
        }
    }

    // reduce min/argmin across the 16 lanes of each half (ties -> lowest index, matches jnp.argmin)
#pragma unroll
    for (int r = 0; r < 8; ++r) {
#pragma unroll
        for (int m = 1; m <= 8; m <<= 1) {
            float ov = __shfl_xor(minval[r], m, 32);
            int   oi = __shfl_xor(minidx[r], m, 32);
            if (ov < minval[r] || (ov == minval[r] && oi < minidx[r])) {
                minval[r] = ov; minidx[r] = oi;
            }
        }
    }
    if (lm == 0) {
        int half = l >> 4;
#pragma unroll
        for (int r = 0; r < 8; ++r) idx_s[w][r + half * 8] = minidx[r];
    }
    __syncthreads();

    // gather winning codebook columns, write quantized output (float4/lane), accumulate sq error
    float sq = 0.f;
    for (int m = 0; m < 16; ++m) {
        const int kstar = idx_s[w][m];
        const size_t nrow = (size_t)b * N_ + rowBase + m;
        float4 xv = *(const float4*)(X + nrow * D_ + 4 * l);
        float4 q;
        q.x = E[((size_t)b * D_ + 4 * l + 0) * K_ + kstar];   // codebook is L2-resident (4.2 MB)
        q.y = E[((size_t)b * D_ + 4 * l + 1) * K_ + kstar];
        q.z = E[((size_t)b * D_ + 4 * l + 2) * K_ + kstar];
        q.w = E[((size_t)b * D_ + 4 * l + 3) * K_ + kstar];
        *(float4*)(out + nrow * D_ + 4 * l) = q;              // straight-through == quantized
        float dx = q.x - xv.x, dy = q.y - xv.y, dz = q.z - xv.z, dw = q.w - xv.w;
        sq += dx * dx + dy * dy + dz * dz + dw * dw;
    }
#pragma unroll
    for (int m = 1; m <= 16; m <<= 1) sq += __shfl_xor(sq, m, 32);
    if (l == 0) wsum[w] = sq;
    __syncthreads();
    if (tid == 0) {
        float t = 0.f;
        for (int i = 0; i < 8; ++i) t += wsum[i];
        partial[blockIdx.x] = t;   // deterministic: fixed slot per workgroup, no atomics
    }
}

// ---------- kernel 3: deterministic loss reduction ----------
__global__ void vq_finish(const float* __restrict__ partial, float* __restrict__ out_loss) {
    float t = 0.f;
    for (int i = 0; i < B_ * (N_ / ROWS_PER_WG); ++i) t += partial[i];
    out_loss[0] = 1.25f * t / (float)((size_t)B_ * N_ * D_);
}

extern "C" void kernel_launch(void* const* d_in, const int* in_sizes, int n_in,
                              void* d_out, int out_size, void* d_ws, size_t ws_size,
                              hipStream_t stream) {
    (void)in_sizes; (void)n_in; (void)out_size; (void)ws_size;
    const float* X = (const float*)d_in[0];   // [B,N,D] fp32
    const float* E = (const float*)d_in[1];   // [B,D,K] fp32
    float* out     = (float*)d_out;           // [B,N,D] quantized_st ++ [1] loss
    float* partial = (float*)d_ws;            // 256 per-workgroup partials
    float* eNorm   = partial + 256;           // 8192 code norms

    const size_t dyn_lds = 2u * TILE_WORDS * sizeof(float);  // ~67 KB (gfx1250: 320 KB/WGP)
    vq_enorm<<<(B_ * K_) / 256, 256, 0, stream>>>(E, eNorm);
    vq_main<<<B_ * (N_ / ROWS_PER_WG), 256, dyn_lds, stream>>>(X, E, eNorm, out, partial);
    vq_finish<<<1, 1, 0, stream>>>(partial, out + (size_t)B_ * N_ * D_);
}